// Discriminator_28767690948703
// MI455X (gfx1250) — compile-verified
//
#include <hip/hip_runtime.h>

typedef __attribute__((ext_vector_type(2))) float v2f;
typedef __attribute__((ext_vector_type(8))) float v8f;

#define LRELU_ALPHA 0.3f
#define LN_EPS 1e-3f

// ---------------------------------------------------------------------------
// Zero-padding copy: src[R][C] -> dst[Rp][Cp], zero fill outside [R,C).
// ---------------------------------------------------------------------------
__global__ __launch_bounds__(256) void pad_copy_kernel(
    const float* __restrict__ src, int R, int C,
    float* __restrict__ dst, int Rp, int Cp)
{
    const int idx = blockIdx.x * 256 + threadIdx.x;
    if (idx >= Rp * Cp) return;
    const int r = idx / Cp, c = idx % Cp;
    dst[idx] = (r < R && c < C) ? src[r * C + c] : 0.f;
}

// ---------------------------------------------------------------------------
// Row-major fp32 GEMM on padded shapes: D[M,Np] = A[M,K](lda) @ B[K,Np] + bias
//   - Np = ldb = ldd, multiple of 32;  K multiple of 4;  M multiple of 16
//   - one wave32 per 16x32 strip: two V_WMMA_F32_16X16X4_F32 per K-step,
//     A fragment (global_load_b64) reused across both accumulators.
//   - branch-free inner loop; EXEC all-1s (only wave-uniform early exit).
// ---------------------------------------------------------------------------
__global__ __launch_bounds__(256) void gemm_wmma_f32_kernel(
    const float* __restrict__ A, int lda,
    const float* __restrict__ B, int ldb,
    const float* __restrict__ bias, int nbias,
    float* __restrict__ D,
    int M, int K, int tilesN2)
{
    const int lane = threadIdx.x & 31;
    const int wib  = threadIdx.x >> 5;                 // 8 waves / 256-thr block
    const int wave = blockIdx.x * 8 + wib;
    if (wave >= (M >> 4) * tilesN2) return;            // wave-uniform

    const int tm = (wave / tilesN2) << 4;              // 16 rows
    const int tn = (wave % tilesN2) << 5;              // 32 cols

    const int half = lane >> 4;                        // 0 -> K{0,1}, 1 -> K{2,3}
    const int l16  = lane & 15;

    const float* Arow  = A + (size_t)(tm + l16) * lda + 2 * half;
    const float* Bc0   = B + (size_t)(2 * half) * ldb + tn + l16;
    const float* Bc1   = Bc0 + 16;
    const size_t bstep = (size_t)4 * ldb;

    v8f acc0 = {0.f,0.f,0.f,0.f,0.f,0.f,0.f,0.f};
    v8f acc1 = {0.f,0.f,0.f,0.f,0.f,0.f,0.f,0.f};

#pragma unroll 4
    for (int k0 = 0; k0 < K; k0 += 4) {
        const v2f a = *(const v2f*)Arow;               // global_load_b64
        v2f b0, b1;
        b0.x = Bc0[0]; b0.y = Bc0[ldb];
        b1.x = Bc1[0]; b1.y = Bc1[ldb];

        acc0 = __builtin_amdgcn_wmma_f32_16x16x4_f32(
            false, a, false, b0, (short)0, acc0, false, false);
        acc1 = __builtin_amdgcn_wmma_f32_16x16x4_f32(
            false, a, false, b1, (short)0, acc1, false, false);

        Arow += 4;
        Bc0  += bstep;
        Bc1  += bstep;
        __builtin_prefetch(Arow + 32, 0, 3);           // global_prefetch_b8
    }

    const int c0 = tn + l16, c1 = c0 + 16;
    const float bv0 = (c0 < nbias) ? bias[c0] : 0.f;
    const float bv1 = (c1 < nbias) ? bias[c1] : 0.f;
#pragma unroll
    for (int r = 0; r < 8; ++r) {
        const int row = tm + r + 8 * half;             // C/D layout: vgpr r -> row r / r+8
        D[(size_t)row * ldb + c0] = acc0[r] + bv0;
        D[(size_t)row * ldb + c1] = acc1[r] + bv1;
    }
}

// ---------------------------------------------------------------------------
// Batch diversity: Mmat[N rows, stride ldm] viewed as [N,10,10].
// div[i,k] = sum_j exp( -sum_d |M[i,k,d] - M[j,k,d]| )
// ---------------------------------------------------------------------------
__global__ __launch_bounds__(128) void diversity_kernel(
    const float* __restrict__ Mmat, int ldm, float* __restrict__ divOut, int Nrows)
{
    __shared__ float rowI[100];
    __shared__ float partial[10][128];

    const int i   = blockIdx.x;
    const int tid = threadIdx.x;

    if (tid < 100) rowI[tid] = Mmat[(size_t)i * ldm + tid];
    __syncthreads();

    float s[10];
#pragma unroll
    for (int k = 0; k < 10; ++k) s[k] = 0.f;

    for (int j = tid; j < Nrows; j += 128) {
        const float* rj = Mmat + (size_t)j * ldm;
#pragma unroll
        for (int k = 0; k < 10; ++k) {
            float l1 = 0.f;
#pragma unroll
            for (int d = 0; d < 10; ++d)
                l1 += fabsf(rowI[k * 10 + d] - rj[k * 10 + d]);
            s[k] += __expf(-l1);
        }
    }

#pragma unroll
    for (int k = 0; k < 10; ++k) partial[k][tid] = s[k];
    __syncthreads();

    if (tid < 10) {
        float t = 0.f;
        for (int u = 0; u < 128; ++u) t += partial[tid][u];
        divOut[(size_t)i * 10 + tid] = t;
    }
}

// ---------------------------------------------------------------------------
// concat([h(256), div(10)]) -> LayerNorm(eps=1e-3, +beta) -> LeakyReLU(0.3)
// Output row stride 272; pad columns 266..271 zeroed for the next GEMM's K.
// ---------------------------------------------------------------------------
__global__ __launch_bounds__(256) void ln_leaky_kernel(
    const float* __restrict__ h, const float* __restrict__ divIn,
    const float* __restrict__ beta, float* __restrict__ cat)
{
    __shared__ float redS[256];
    __shared__ float redQ[256];

    const int i   = blockIdx.x;
    const int tid = threadIdx.x;

    const float x0 = h[(size_t)i * 256 + tid];
    const float x1 = (tid < 10) ? divIn[(size_t)i * 10 + tid] : 0.f;

    redS[tid] = x0 + x1;
    redQ[tid] = x0 * x0 + x1 * x1;
    __syncthreads();
    for (int off = 128; off >= 1; off >>= 1) {
        if (tid < off) { redS[tid] += redS[tid + off]; redQ[tid] += redQ[tid + off]; }
        __syncthreads();
    }

    const float inv  = 1.f / 266.f;
    const float mean = redS[0] * inv;
    const float var  = redQ[0] * inv - mean * mean;   // jnp.var (ddof=0)
    const float rstd = rsqrtf(var + LN_EPS);

    float y0 = (x0 - mean) * rstd + beta[tid];
    y0 = (y0 >= 0.f) ? y0 : LRELU_ALPHA * y0;
    cat[(size_t)i * 272 + tid] = y0;

    if (tid < 10) {
        float y1 = (x1 - mean) * rstd + beta[256 + tid];
        y1 = (y1 >= 0.f) ? y1 : LRELU_ALPHA * y1;
        cat[(size_t)i * 272 + 256 + tid] = y1;
    }
    if (tid < 6) cat[(size_t)i * 272 + 266 + tid] = 0.f;   // zero K-padding
}

// ---------------------------------------------------------------------------
// Critic head: out[i] = dot(cat[i, 0:266], Wf) + bf   (cat row stride 272)
// ---------------------------------------------------------------------------
__global__ __launch_bounds__(256) void head_kernel(
    const float* __restrict__ cat, const float* __restrict__ Wf,
    const float* __restrict__ bf, float* __restrict__ out)
{
    __shared__ float red[256];
    const int i   = blockIdx.x;
    const int tid = threadIdx.x;

    float s = 0.f;
    for (int c = tid; c < 266; c += 256)
        s += cat[(size_t)i * 272 + c] * Wf[c];
    red[tid] = s;
    __syncthreads();
    for (int off = 128; off >= 1; off >>= 1) {
        if (tid < off) red[tid] += red[tid + off];
        __syncthreads();
    }
    if (tid == 0) out[i] = red[0] + bf[0];
}

// ---------------------------------------------------------------------------
// Launch pipeline (all on `stream`, graph-capture safe, no host syncs)
// ---------------------------------------------------------------------------
extern "C" void kernel_launch(void* const* d_in, const int* in_sizes, int n_in,
                              void* d_out, int out_size, void* d_ws, size_t ws_size,
                              hipStream_t stream) {
    (void)in_sizes; (void)n_in; (void)out_size; (void)ws_size;

    const int N = 1024, NFEAT = 512, HID = 256, MB = 100;
    const int MBP = 128;        // padded minibatch-disc width (mult of 32)
    const int CATP = 272;       // padded concat width (mult of 4)

    const float* x     = (const float*)d_in[0];
    const float* W0    = (const float*)d_in[1];
    const float* b0    = (const float*)d_in[2];
    const float* Wd0   = (const float*)d_in[3];
    const float* bd0   = (const float*)d_in[4];
    const float* beta0 = (const float*)d_in[5];
    const float* W1    = (const float*)d_in[6];
    const float* b1    = (const float*)d_in[7];
    const float* Wd1   = (const float*)d_in[8];
    const float* bd1   = (const float*)d_in[9];
    const float* beta1 = (const float*)d_in[10];
    const float* Wf    = (const float*)d_in[11];
    const float* bf    = (const float*)d_in[12];
    float* out = (float*)d_out;

    float* ws     = (float*)d_ws;
    float* hBuf   = ws;                               // 1024*256
    float* mBuf   = hBuf   + (size_t)N * HID;         // 1024*128
    float* divBuf = mBuf   + (size_t)N * MBP;         // 1024*10
    float* catBuf = divBuf + (size_t)N * 10;          // 1024*272
    float* W1p    = catBuf + (size_t)N * CATP;        // 272*256
    float* Wd0p   = W1p    + (size_t)CATP * HID;      // 256*128
    float* Wd1p   = Wd0p   + (size_t)HID * MBP;       // 256*128

    // ---- Build zero-padded weight copies (regularize all GEMM shapes) ----
    pad_copy_kernel<<<(CATP * HID + 255) / 256, 256, 0, stream>>>(W1, 266, HID, W1p, CATP, HID);
    pad_copy_kernel<<<(HID * MBP + 255) / 256, 256, 0, stream>>>(Wd0, HID, MB, Wd0p, HID, MBP);
    pad_copy_kernel<<<(HID * MBP + 255) / 256, 256, 0, stream>>>(Wd1, HID, MB, Wd1p, HID, MBP);

    auto gemmBlocks = [](int M, int tilesN2) {
        int waves = (M >> 4) * tilesN2;
        return (waves + 7) / 8;                       // 8 waves (256 thr) / block
    };

    // ---- Block 0 ----
    gemm_wmma_f32_kernel<<<gemmBlocks(N, HID / 32), 256, 0, stream>>>(
        x, NFEAT, W0, HID, b0, HID, hBuf, N, NFEAT, HID / 32);
    gemm_wmma_f32_kernel<<<gemmBlocks(N, MBP / 32), 256, 0, stream>>>(
        hBuf, HID, Wd0p, MBP, bd0, MB, mBuf, N, HID, MBP / 32);
    diversity_kernel<<<N, 128, 0, stream>>>(mBuf, MBP, divBuf, N);
    ln_leaky_kernel<<<N, 256, 0, stream>>>(hBuf, divBuf, beta0, catBuf);

    // ---- Block 1 (K = 272 padded concat) ----
    gemm_wmma_f32_kernel<<<gemmBlocks(N, HID / 32), 256, 0, stream>>>(
        catBuf, CATP, W1p, HID, b1, HID, hBuf, N, CATP, HID / 32);
    gemm_wmma_f32_kernel<<<gemmBlocks(N, MBP / 32), 256, 0, stream>>>(
        hBuf, HID, Wd1p, MBP, bd1, MB, mBuf, N, HID, MBP / 32);
    diversity_kernel<<<N, 128, 0, stream>>>(mBuf, MBP, divBuf, N);
    ln_leaky_kernel<<<N, 256, 0, stream>>>(hBuf, divBuf, beta1, catBuf);

    // ---- Critic head ----
    head_kernel<<<N, 256, 0, stream>>>(catBuf, Wf, bf, out);
}